// JKGAT_26877905338522
// MI455X (gfx1250) — compile-verified
//
#include <hip/hip_runtime.h>
#include <math.h>

// Problem constants (match reference)
#define GAT_H   4
#define GAT_D   32
#define GAT_HD  128      // H*D
#define GAT_LHD 384      // L*H*D
#define GAT_C   32
#define SLOPE   0.2f

typedef __attribute__((ext_vector_type(2))) float v2f;
typedef __attribute__((ext_vector_type(8))) float v8f;

// ---------------------------------------------------------------------------
// Generic fp32 WMMA GEMM: Y[nrows x NC] = X[nrows x K] (row stride ldx) @ W[K x NC]
// Y has row stride NC (compile-time), so store offsets fold into immediates.
// 4 waves / block, each wave computes a 16-row strip across all NC columns.
// Uses V_WMMA_F32_16X16X4_F32 (fp32 in/out, matches reference numerics).
//
// W is staged in LDS in a K-pair-interleaved layout:
//   Wlds[(kk & ~1)*NC + 2*col + (kk & 1)] = W[kk*NC + col]
// so each B operand (two K-adjacent values at one column) is one ds_load_b64
// landing directly in an even-aligned VGPR pair for WMMA.
// ---------------------------------------------------------------------------
template <int K, int NC>
__global__ __launch_bounds__(128) void gemm_wmma_f32(
    const float* __restrict__ X, int ldx,
    const float* __restrict__ W,
    float* __restrict__ Y, int nrows)
{
    constexpr int NT = NC / 16;
    __shared__ __align__(16) float Wlds[K * NC];
    for (int i = threadIdx.x; i < K * NC; i += blockDim.x) {
        int kk  = i / NC;
        int col = i - kk * NC;
        Wlds[(kk & ~1) * NC + 2 * col + (kk & 1)] = W[i];
    }
    __syncthreads();

    const int wave    = threadIdx.x >> 5;
    const int lane    = threadIdx.x & 31;
    const int lane16  = lane & 15;
    const int kq      = lane >> 4;          // 0 for lanes 0-15, 1 for 16-31
    const int rowbase = blockIdx.x * 64 + wave * 16;

    v8f acc[NT];
    #pragma unroll
    for (int t = 0; t < NT; ++t)
        #pragma unroll
        for (int i = 0; i < 8; ++i) acc[t][i] = 0.0f;

    // Clamp instead of predicate: garbage A rows only produce garbage D rows,
    // which are masked at store time. Avoids divergent loads in the k-loop.
    int arow = rowbase + lane16;
    if (arow >= nrows) arow = nrows - 1;
    const float* xrow = X + (size_t)arow * ldx;

    #pragma unroll 2
    for (int k = 0; k < K; k += 4) {
        // A: 16x4 fp32 tile. VGPR0 = {K=k | K=k+2}, VGPR1 = {K=k+1 | K=k+3}
        v2f a = *(const v2f*)(xrow + k + 2 * kq);
        const int pairRow = (k + 2 * kq) * NC;   // even kk pair base
        #pragma unroll
        for (int t = 0; t < NT; ++t) {
            v2f b = *(const v2f*)(Wlds + pairRow + 2 * (t * 16 + lane16));
            acc[t] = __builtin_amdgcn_wmma_f32_16x16x4_f32(
                false, a, false, b, (short)0, acc[t], false, false);
        }
    }

    // D layout: VGPR v -> row (v | v+8), col = lane16.
    // Base offset per lane; (v,t) offsets are compile-time constants.
    const int half = (lane >= 16) ? 8 : 0;
    float* ybase = Y + (rowbase + half) * NC + lane16;
    if (rowbase + 16 <= nrows) {
        // full strip: unguarded stores with immediate offsets
        #pragma unroll
        for (int t = 0; t < NT; ++t)
            #pragma unroll
            for (int v = 0; v < 8; ++v)
                ybase[v * NC + t * 16] = acc[t][v];
    } else {
        // tail block: guard once per row, covering all column tiles
        #pragma unroll
        for (int v = 0; v < 8; ++v) {
            if (rowbase + half + v < nrows) {
                #pragma unroll
                for (int t = 0; t < NT; ++t)
                    ybase[v * NC + t * 16] = acc[t][v];
            }
        }
    }
}

// ---------------------------------------------------------------------------
// el[n,h] = sum_d feat[n,h,d]*al[h,d] ; er likewise
// ---------------------------------------------------------------------------
__global__ __launch_bounds__(256) void attn_coef_kernel(
    const float* __restrict__ feat, const float* __restrict__ al,
    const float* __restrict__ ar, float* __restrict__ el,
    float* __restrict__ er, int n)
{
    int tid = blockIdx.x * blockDim.x + threadIdx.x;
    if (tid >= n * GAT_H) return;
    int node = tid >> 2;
    int h    = tid & 3;
    const float* f = feat + (size_t)node * GAT_HD + h * GAT_D;
    const float* a = al + h * GAT_D;
    const float* b = ar + h * GAT_D;
    float sl = 0.f, sr = 0.f;
    #pragma unroll 8
    for (int d = 0; d < GAT_D; ++d) {
        float fv = f[d];
        sl += fv * a[d];
        sr += fv * b[d];
    }
    el[tid] = sl;
    er[tid] = sr;
}

// ---------------------------------------------------------------------------
// init: m = -inf, s = 0, agg = 0
// ---------------------------------------------------------------------------
__global__ __launch_bounds__(256) void init_kernel(
    float* __restrict__ m, float* __restrict__ s,
    float* __restrict__ agg, int nh, int nfeat)
{
    int tid = blockIdx.x * blockDim.x + threadIdx.x;
    if (tid < nh) { m[tid] = -INFINITY; s[tid] = 0.f; }
    if (tid < nfeat) agg[tid] = 0.f;
}

__device__ __forceinline__ void atomicMaxFloat(float* addr, float val)
{
    if (val >= 0.f)
        atomicMax((int*)addr, __float_as_int(val));
    else
        atomicMin((unsigned int*)addr, __float_as_uint(val));
}

// ---------------------------------------------------------------------------
// Pass 1: e = leaky_relu(el[src]+er[dst]); ebuf <- e; m[dst,h] = max
// ---------------------------------------------------------------------------
__global__ __launch_bounds__(256) void edge_score_kernel(
    const int* __restrict__ src, const int* __restrict__ dst,
    const float* __restrict__ el, const float* __restrict__ er,
    float* __restrict__ ebuf, float* __restrict__ m, int ne)
{
    int e = blockIdx.x * blockDim.x + threadIdx.x;
    if (e >= ne) return;
    int s = src[e], d = dst[e];
    #pragma unroll
    for (int h = 0; h < GAT_H; ++h) {
        float v = el[s * GAT_H + h] + er[d * GAT_H + h];
        v = (v > 0.f) ? v : SLOPE * v;
        ebuf[e * GAT_H + h] = v;
        atomicMaxFloat(&m[d * GAT_H + h], v);
    }
}

// ---------------------------------------------------------------------------
// Pass 2: ex = exp(e - m[dst]); ebuf <- ex; s[dst,h] += ex
// ---------------------------------------------------------------------------
__global__ __launch_bounds__(256) void edge_exp_kernel(
    const int* __restrict__ dst, const float* __restrict__ m,
    float* __restrict__ ebuf, float* __restrict__ ssum, int ne)
{
    int e = blockIdx.x * blockDim.x + threadIdx.x;
    if (e >= ne) return;
    int d = dst[e];
    #pragma unroll
    for (int h = 0; h < GAT_H; ++h) {
        float ex = __expf(ebuf[e * GAT_H + h] - m[d * GAT_H + h]);
        ebuf[e * GAT_H + h] = ex;
        atomicAdd(&ssum[d * GAT_H + h], ex);
    }
}

// ---------------------------------------------------------------------------
// Pass 3: agg[dst,h,:] += feat[src,h,:] * (ex / s[dst,h])
// one thread per (edge, head, 4-dword chunk) -> float4 gather + 4 atomics
// ---------------------------------------------------------------------------
__global__ __launch_bounds__(256) void edge_aggregate_kernel(
    const int* __restrict__ src, const int* __restrict__ dst,
    const float* __restrict__ ebuf, const float* __restrict__ ssum,
    const float* __restrict__ feat, float* __restrict__ agg, int ne)
{
    int tid = blockIdx.x * blockDim.x + threadIdx.x;
    int e = tid >> 5;                 // 32 threads per edge (4 heads * 8 chunks)
    if (e >= ne) return;
    int r  = tid & 31;
    int h  = r >> 3;
    int dq = (r & 7) * 4;
    int s  = src[e], d = dst[e];
    float a = ebuf[e * GAT_H + h] / ssum[d * GAT_H + h];
    const float4 f = *(const float4*)(feat + (size_t)s * GAT_HD + h * GAT_D + dq);
    float* o = agg + (size_t)d * GAT_HD + h * GAT_D + dq;
    atomicAdd(o + 0, f.x * a);
    atomicAdd(o + 1, f.y * a);
    atomicAdd(o + 2, f.z * a);
    atomicAdd(o + 3, f.w * a);
}

// ---------------------------------------------------------------------------
// ELU and write into the JK-concat slice (row stride ldo)
// ---------------------------------------------------------------------------
__global__ __launch_bounds__(256) void elu_kernel(
    const float* __restrict__ agg, float* __restrict__ out, int ldo, int n)
{
    int tid = blockIdx.x * blockDim.x + threadIdx.x;
    if (tid >= n * GAT_HD) return;
    int node = tid >> 7;
    int c    = tid & 127;
    float v = agg[tid];
    v = (v > 0.f) ? v : (__expf(v) - 1.f);
    out[(size_t)node * ldo + c] = v;
}

// ---------------------------------------------------------------------------
extern "C" void kernel_launch(void* const* d_in, const int* in_sizes, int n_in,
                              void* d_out, int out_size, void* d_ws, size_t ws_size,
                              hipStream_t stream)
{
    const float* x    = (const float*)d_in[0];
    const int*   src  = (const int*)d_in[1];
    const int*   dst  = (const int*)d_in[2];
    const float* Wl[3]  = { (const float*)d_in[3], (const float*)d_in[6], (const float*)d_in[9] };
    const float* all[3] = { (const float*)d_in[4], (const float*)d_in[7], (const float*)d_in[10] };
    const float* arl[3] = { (const float*)d_in[5], (const float*)d_in[8], (const float*)d_in[11] };
    const float* Wproj  = (const float*)d_in[12];
    float* logits = (float*)d_out;

    const int n  = in_sizes[0] / 128;   // 50000
    const int ne = in_sizes[1];         // 800000

    // workspace layout (floats)
    float* p    = (float*)d_ws;
    float* emb  = p;                 p += (size_t)n * GAT_LHD;  // [N, 384]
    float* feat = p;                 p += (size_t)n * GAT_HD;   // [N, 128]
    float* agg  = p;                 p += (size_t)n * GAT_HD;   // [N, 128]
    float* el   = p;                 p += (size_t)n * GAT_H;
    float* er   = p;                 p += (size_t)n * GAT_H;
    float* m    = p;                 p += (size_t)n * GAT_H;
    float* ss   = p;                 p += (size_t)n * GAT_H;
    float* ebuf = p;                 p += (size_t)ne * GAT_H;

    const int gemm_grid = (n + 63) / 64;
    const int nh_grid   = (n * GAT_H + 255) / 256;
    const int nf_grid   = (n * GAT_HD + 255) / 256;
    const int e_grid    = (ne + 255) / 256;
    const int ea_grid   = (ne * 32 + 255) / 256;

    for (int l = 0; l < 3; ++l) {
        // feat = h @ W   (layer0 reads x with ldx=128, others read emb slices)
        if (l == 0)
            gemm_wmma_f32<128, GAT_HD><<<gemm_grid, 128, 0, stream>>>(
                x, 128, Wl[0], feat, n);
        else
            gemm_wmma_f32<128, GAT_HD><<<gemm_grid, 128, 0, stream>>>(
                emb + (size_t)(l - 1) * GAT_HD, GAT_LHD, Wl[l], feat, n);

        attn_coef_kernel<<<nh_grid, 256, 0, stream>>>(feat, all[l], arl[l], el, er, n);
        init_kernel<<<nf_grid, 256, 0, stream>>>(m, ss, agg, n * GAT_H, n * GAT_HD);
        edge_score_kernel<<<e_grid, 256, 0, stream>>>(src, dst, el, er, ebuf, m, ne);
        edge_exp_kernel<<<e_grid, 256, 0, stream>>>(dst, m, ebuf, ss, ne);
        edge_aggregate_kernel<<<ea_grid, 256, 0, stream>>>(src, dst, ebuf, ss, feat, agg, ne);
        elu_kernel<<<nf_grid, 256, 0, stream>>>(agg, emb + (size_t)l * GAT_HD, GAT_LHD, n);
    }

    // logits = emb @ Wproj   [N,384] x [384,32]
    gemm_wmma_f32<GAT_LHD, GAT_C><<<gemm_grid, 128, 0, stream>>>(
        emb, GAT_LHD, Wproj, logits, n);
}